// SimplexProj_34694745817328
// MI455X (gfx1250) — compile-verified
//
#include <hip/hip_runtime.h>
#include <cstdint>
#include <cstddef>

// Simplex projection (Z = 1) along rows of a 4096 x 12288 f32 matrix.
// One workgroup per row. Row is staged into LDS with CDNA5 async
// global->LDS B128 copies (ASYNCcnt, non-temporal), then Michelot's
// sort-free algorithm finds the shift l with a few masked-sum sweeps over
// LDS, and the final pass streams wp = max(x+l,0) and wc = x-wp back to
// HBM with non-temporal stores.

typedef float v4f __attribute__((ext_vector_type(4)));  // native vector: b128 ld/st

constexpr int COLS   = 12288;
constexpr int VEC4   = COLS / 4;     // 3072 float4 per row
constexpr int TPB    = 256;          // 8 waves (wave32)
constexpr int CHUNKS = VEC4 / TPB;   // 12 float4 per thread

__global__ __launch_bounds__(TPB) void simplex_proj_kernel(
    const float* __restrict__ x, float* __restrict__ out, int rows)
{
    __shared__ v4f   sdata[VEC4];       // 48 KB row staging
    __shared__ float red[18];           // 8 wave partials (s,c) + broadcast (l, C)

    const int t   = threadIdx.x;
    const int row = blockIdx.x;

    // ---- Async copy: global row -> LDS (CDNA5 GLOBAL_LOAD_ASYNC_TO_LDS_B128) ----
    // Row is read exactly once -> non-temporal load hint to keep L2 clean.
    const unsigned long long gbase =
        (unsigned long long)(uintptr_t)(x + (size_t)row * COLS);
    const unsigned int lds_base = (unsigned int)(uintptr_t)(void*)&sdata[0];
#pragma unroll
    for (int k = 0; k < CHUNKS; ++k) {
        unsigned int boff  = (unsigned int)((t + k * TPB) * 16); // byte offset in row
        unsigned int laddr = lds_base + boff;                    // LDS byte address
        // GVS mode: vdst = LDS addr VGPR, vaddr = 32-bit offset, saddr = 64-bit base
        asm volatile("global_load_async_to_lds_b128 %0, %1, %2 th:TH_LOAD_NT"
                     :: "v"(laddr), "v"(boff), "s"(gbase)
                     : "memory");
    }
    asm volatile("s_wait_asynccnt 0x0" ::: "memory"); // my async writes to LDS done
    __syncthreads();                                   // everyone's writes visible

    // ---- Michelot iteration: find l s.t. sum(max(x + l, 0)) == 1 ----
    const int wid  = t >> 5;
    const int lane = t & 31;

    float l     = 3.0e38f;  // first sweep: everything active -> total sum & n
    float prevC = -1.0f;
    float lfin  = 0.0f;

    for (int it = 0; it < 64; ++it) {
        float s = 0.0f, c = 0.0f;
#pragma unroll
        for (int k = 0; k < CHUNKS; ++k) {
            v4f v = sdata[t + k * TPB];
            if (v.x + l > 0.0f) { s += v.x; c += 1.0f; }
            if (v.y + l > 0.0f) { s += v.y; c += 1.0f; }
            if (v.z + l > 0.0f) { s += v.z; c += 1.0f; }
            if (v.w + l > 0.0f) { s += v.w; c += 1.0f; }
        }
        // fixed-order wave32 reduction
#pragma unroll
        for (int off = 16; off > 0; off >>= 1) {
            s += __shfl_xor(s, off, 32);
            c += __shfl_xor(c, off, 32);
        }
        if (lane == 0) { red[2 * wid] = s; red[2 * wid + 1] = c; }
        __syncthreads();
        if (t == 0) {
            float S = 0.0f, C = 0.0f;
            for (int w = 0; w < 8; ++w) { S += red[2 * w]; C += red[2 * w + 1]; }
            red[16] = (1.0f - S) / C;   // new shift l
            red[17] = C;                // active count
        }
        __syncthreads();
        const float nl = red[16];
        const float C  = red[17];
        __syncthreads();               // protect red[] before next iteration's writes

        lfin = nl;
        if (C == prevC || C < 1.0f) break;  // active set stable -> converged
        prevC = C;
        l     = nl;
    }

    // ---- Final pass: wp = max(x + l, 0), wc = x - wp (streaming NT stores) ----
    v4f* __restrict__ wp = (v4f*)out + (size_t)row * VEC4;
    v4f* __restrict__ wc = (v4f*)out + (size_t)rows * VEC4 + (size_t)row * VEC4;
#pragma unroll
    for (int k = 0; k < CHUNKS; ++k) {
        const int i = t + k * TPB;
        v4f v = sdata[i];
        v4f p, q;
        p.x = fmaxf(v.x + lfin, 0.0f); q.x = v.x - p.x;
        p.y = fmaxf(v.y + lfin, 0.0f); q.y = v.y - p.y;
        p.z = fmaxf(v.z + lfin, 0.0f); q.z = v.z - p.z;
        p.w = fmaxf(v.w + lfin, 0.0f); q.w = v.w - p.w;
        __builtin_nontemporal_store(p, &wp[i]);
        __builtin_nontemporal_store(q, &wc[i]);
    }
}

extern "C" void kernel_launch(void* const* d_in, const int* in_sizes, int n_in,
                              void* d_out, int out_size, void* d_ws, size_t ws_size,
                              hipStream_t stream)
{
    const float* x = (const float*)d_in[0];
    float* out = (float*)d_out;
    const int rows = in_sizes[0] / COLS;   // 4096
    simplex_proj_kernel<<<rows, TPB, 0, stream>>>(x, out, rows);
}